// EdgeDecoder_9320079033006
// MI455X (gfx1250) — compile-verified
//
#include <hip/hip_runtime.h>

typedef float v2f __attribute__((ext_vector_type(2)));
typedef float v8f __attribute__((ext_vector_type(8)));

#define HDIM 128   // H
#define TILE_M 16  // edges per wave tile

__global__ __launch_bounds__(256) void edge_decoder_wmma(
    const float* __restrict__ zu,   // [N_USER, 128]
    const float* __restrict__ zm,   // [N_MOVIE, 128]
    const int*   __restrict__ row,  // [E]
    const int*   __restrict__ col,  // [E]
    const float* __restrict__ W1,   // [256, 128] row-major (k, n)
    const float* __restrict__ b1,   // [128]
    const float* __restrict__ W2,   // [128, 3]
    const float* __restrict__ b2,   // [3]
    float* __restrict__ out,        // [E, 3]
    int E, int nTiles)
{
    // 64 KB: half of W1, swizzled so sB[(k>>1)*256 + 2n + (k&1)] holds W1[k][n]
    // => B-fragment (W1[kk][n], W1[kk+1][n]) is one 8B-aligned ds_load_b64.
    __shared__ float sB[16384];

    const int tid  = threadIdx.x;
    const int lane = tid & 31;
    const int waveInBlk = tid >> 5;
    const int tile = blockIdx.x * 8 + waveInBlk;
    const bool valid = (tile < nTiles);

    const int half = lane >> 4;   // K sub-group of the fragment
    const int l15  = lane & 15;   // M index (A) / N index (B, C)

    // Per-lane gathered A-row base pointers (this lane's edge row, K-phase offset)
    const int eBase = tile * TILE_M;
    int eClamped = eBase + l15;
    if (eClamped >= E) eClamped = E > 0 ? E - 1 : 0;
    const long long ru = valid ? (long long)row[eClamped] : 0ll;
    const long long cm = valid ? (long long)col[eClamped] : 0ll;
    const float* __restrict__ pu = zu + ru * HDIM + 2 * half;
    const float* __restrict__ pm = zm + cm * HDIM + 2 * half;

    v8f acc[8];
    #pragma unroll
    for (int t = 0; t < 8; ++t) acc[t] = v8f{};

    // ---------------- phase 0: W1 rows k = 0..127  (z_user part) -------------
    {
        const float4* g = (const float4*)(W1);           // 4096 float4s
        for (int i = tid; i < 4096; i += 256) {
            float4 v = g[i];
            int k = i >> 5;                 // (4i)/128
            int n = (i << 2) & 127;         // (4i)%128
            int base = (k >> 1) * 256 + (k & 1) + 2 * n;
            sB[base + 0] = v.x; sB[base + 2] = v.y;
            sB[base + 4] = v.z; sB[base + 6] = v.w;
        }
    }
    __syncthreads();
    if (valid) {
        #pragma unroll 4
        for (int k0 = 0; k0 < 128; k0 += 4) {
            v2f a = *(const v2f*)(pu + k0);
            const float* bp = sB + (k0 >> 1) * 256 + half * 256 + 2 * l15;
            #pragma unroll
            for (int nt = 0; nt < 8; ++nt) {
                v2f b = *(const v2f*)(bp + 32 * nt);
                acc[nt] = __builtin_amdgcn_wmma_f32_16x16x4_f32(
                    false, a, false, b, (short)0, acc[nt], false, false);
            }
        }
    }
    __syncthreads();
    // ---------------- phase 1: W1 rows k = 128..255 (z_movie part) -----------
    {
        const float4* g = (const float4*)(W1 + 128 * HDIM);
        for (int i = tid; i < 4096; i += 256) {
            float4 v = g[i];
            int k = i >> 5;
            int n = (i << 2) & 127;
            int base = (k >> 1) * 256 + (k & 1) + 2 * n;
            sB[base + 0] = v.x; sB[base + 2] = v.y;
            sB[base + 4] = v.z; sB[base + 6] = v.w;
        }
    }
    __syncthreads();
    if (valid) {
        #pragma unroll 4
        for (int k0 = 0; k0 < 128; k0 += 4) {
            v2f a = *(const v2f*)(pm + k0);
            const float* bp = sB + (k0 >> 1) * 256 + half * 256 + 2 * l15;
            #pragma unroll
            for (int nt = 0; nt < 8; ++nt) {
                v2f b = *(const v2f*)(bp + 32 * nt);
                acc[nt] = __builtin_amdgcn_wmma_f32_16x16x4_f32(
                    false, a, false, b, (short)0, acc[nt], false, false);
            }
        }

        // -------- epilogue: bias + relu, then h @ W2 + b2 (N = 3) in VALU ----
        // C layout: acc[nt][v] = h[M = v + 8*half][N = l15 + 16*nt]
        float p0[8], p1[8], p2[8];
        #pragma unroll
        for (int v = 0; v < 8; ++v) { p0[v] = 0.f; p1[v] = 0.f; p2[v] = 0.f; }

        #pragma unroll
        for (int nt = 0; nt < 8; ++nt) {
            const int n = l15 + 16 * nt;
            const float bias = b1[n];
            const float w20 = W2[n * 3 + 0];
            const float w21 = W2[n * 3 + 1];
            const float w22 = W2[n * 3 + 2];
            #pragma unroll
            for (int v = 0; v < 8; ++v) {
                float h = acc[nt][v] + bias;
                h = fmaxf(h, 0.0f);
                p0[v] = fmaf(h, w20, p0[v]);
                p1[v] = fmaf(h, w21, p1[v]);
                p2[v] = fmaf(h, w22, p2[v]);
            }
        }
        // reduce over the 16 lanes of each half-wave (n-dimension)
        #pragma unroll
        for (int off = 1; off < 16; off <<= 1) {
            #pragma unroll
            for (int v = 0; v < 8; ++v) {
                p0[v] += __shfl_xor(p0[v], off, 32);
                p1[v] += __shfl_xor(p1[v], off, 32);
                p2[v] += __shfl_xor(p2[v], off, 32);
            }
        }
        if (l15 == 0) {
            const float bb0 = b2[0], bb1 = b2[1], bb2 = b2[2];
            #pragma unroll
            for (int v = 0; v < 8; ++v) {
                const int e = eBase + v + 8 * half;
                if (e < E) {
                    out[(long long)e * 3 + 0] = p0[v] + bb0;
                    out[(long long)e * 3 + 1] = p1[v] + bb1;
                    out[(long long)e * 3 + 2] = p2[v] + bb2;
                }
            }
        }
    }
}

extern "C" void kernel_launch(void* const* d_in, const int* in_sizes, int n_in,
                              void* d_out, int out_size, void* d_ws, size_t ws_size,
                              hipStream_t stream) {
    const float* zu = (const float*)d_in[0];
    const float* zm = (const float*)d_in[1];
    const int*   row = (const int*)d_in[2];
    const int*   col = (const int*)d_in[3];
    const float* W1 = (const float*)d_in[4];
    const float* b1 = (const float*)d_in[5];
    const float* W2 = (const float*)d_in[6];
    const float* b2 = (const float*)d_in[7];
    float* out = (float*)d_out;

    const int E = in_sizes[2];
    const int nTiles = (E + TILE_M - 1) / TILE_M;
    const int blocks = (nTiles + 7) / 8;     // 8 waves (tiles) per 256-thread block
    if (blocks <= 0) return;

    edge_decoder_wmma<<<blocks, 256, 0, stream>>>(zu, zm, row, col, W1, b1, W2, b2,
                                                  out, E, nTiles);
}